// EarthMoversDistance_43800076485250
// MI455X (gfx1250) — compile-verified
//
#include <hip/hip_runtime.h>

#define B_    16
#define D_    3
#define N_    2048
#define ITERS 100
#define INVN  (1.0f / 2048.0f)
// NEG_SCALE = -(1/eps)*log2(e), eps = 0.1 :  exp(-t/eps) = exp2(t * NEG_SCALE)
#define NEG_SCALE (-14.4269504088896340736f)
// M2POS = -2*NEG_SCALE   (NEG_SCALE*t = NEG_SCALE*(p2+q2) + M2POS*dot)
#define M2POS (28.8539008177792681472f)

typedef __attribute__((ext_vector_type(2))) float v2f;
typedef __attribute__((ext_vector_type(8))) float v8f;

// ---------------------------------------------------------------------------
// Setup: row norms x2,y2; u=v=1/N; zero the scalar output (graph-replay safe).
// ---------------------------------------------------------------------------
__global__ __launch_bounds__(256) void emd_setup(
    const float* __restrict__ x, const float* __restrict__ y,
    float* __restrict__ x2, float* __restrict__ y2,
    float* __restrict__ u, float* __restrict__ v, float* __restrict__ out)
{
    int i = blockIdx.x * 256 + threadIdx.x;      // 0 .. B*N-1
    int b = i >> 11;                             // / N_
    int n = i & (N_ - 1);
    const float* xb = x + (size_t)b * D_ * N_ + n;
    const float* yb = y + (size_t)b * D_ * N_ + n;
    float sx = 0.f, sy = 0.f;
#pragma unroll
    for (int d = 0; d < D_; ++d) {
        float a = xb[d * N_]; sx += a * a;
        float c = yb[d * N_]; sy += c * c;
    }
    x2[i] = sx; y2[i] = sy; u[i] = INVN; v[i] = INVN;
    if (i == 0) out[0] = 0.f;
}

// ---------------------------------------------------------------------------
// Stage Q panel into LDS as K-dim pairs for the WMMA B-tile:
//   qpan2[0][col] = {y0, y1}
//   qpan2[1][col] = {y2, q2scale * q2[col]}   <- norm term rides in K=3 slot
// One branchless ds_load_b64 per chunk fetches a full B column-pair.
// ---------------------------------------------------------------------------
__device__ __forceinline__ void stage_panel(
    const float* __restrict__ Qb, const float* __restrict__ q2b, float q2scale,
    const float* __restrict__ wb, float2 (*qpan2)[N_], float* wls, int tid)
{
    for (int c = tid * 4; c < N_; c += 256 * 4) {
        float4 r0 = *(const float4*)(Qb + 0 * N_ + c);
        float4 r1 = *(const float4*)(Qb + 1 * N_ + c);
        float4 r2 = *(const float4*)(Qb + 2 * N_ + c);
        float4 rq = *(const float4*)(q2b + c);
        qpan2[0][c + 0] = make_float2(r0.x, r1.x);
        qpan2[0][c + 1] = make_float2(r0.y, r1.y);
        qpan2[0][c + 2] = make_float2(r0.z, r1.z);
        qpan2[0][c + 3] = make_float2(r0.w, r1.w);
        qpan2[1][c + 0] = make_float2(r2.x, q2scale * rq.x);
        qpan2[1][c + 1] = make_float2(r2.y, q2scale * rq.y);
        qpan2[1][c + 2] = make_float2(r2.z, q2scale * rq.z);
        qpan2[1][c + 3] = make_float2(r2.w, q2scale * rq.w);
        *(float4*)(wls + c) = *(const float4*)(wb + c);
    }
}

// ---------------------------------------------------------------------------
// One Sinkhorn half-step:  wout[b,i] = invN / ( sum_j K(i,j) * win[b,j] + 1e-8 )
// K(i,j) = exp2(min(s,0)),  s computed ENTIRELY by one WMMA:
//   A = [M2POS*x0, M2POS*x1, M2POS*x2, 1],  B = [y0, y1, y2, NEG_SCALE*q2],
//   C = NEG_SCALE*p2  (loop-invariant register vector)  =>  D = s.
// Grid: B*32 blocks, 256 threads. 4 row-tiles per block; each tile owned by a
// pair of waves interleaving the 128 column chunks, combined through LDS.
// ---------------------------------------------------------------------------
__global__ __launch_bounds__(256) void sinkhorn_matvec(
    const float* __restrict__ P,  const float* __restrict__ Q,
    const float* __restrict__ p2, const float* __restrict__ q2,
    const float* __restrict__ win, float* __restrict__ wout)
{
    __shared__ float2 qpan2[2][N_];   // 32 KB
    __shared__ float  wls[N_];        // 8 KB
    __shared__ float  part[4][8][32]; // 4 KB pair-combine buffer

    const int b   = blockIdx.x >> 5;       // 32 blocks per batch
    const int grp = blockIdx.x & 31;
    const int tid = threadIdx.x;

    stage_panel(Q + (size_t)b * D_ * N_, q2 + (size_t)b * N_, NEG_SCALE,
                win + (size_t)b * N_, qpan2, wls, tid);
    __syncthreads();

    const int wave = tid >> 5;
    const int lane = tid & 31;
    const int half = lane >> 4;          // 0: lanes 0-15, 1: lanes 16-31
    const int l16  = lane & 15;
    const int tile = wave >> 1;          // 4 row-tiles per block
    const int par  = wave & 1;           // column-chunk parity within the pair
    const int i0   = (grp * 4 + tile) * 16;

    // A tile (16x4 f32): lanes 0-15 hold K=0,1; lanes 16-31 hold K=2,3.
    const float* Pb = P + (size_t)b * D_ * N_;
    const int arow = i0 + l16;
    v2f A;
    A.x = M2POS * Pb[(2 * half) * N_ + arow];
    A.y = (half == 0) ? (M2POS * Pb[1 * N_ + arow]) : 1.0f;  // K=3 -> 1

    // loop-invariant C operand: NEG_SCALE * p2 for rows r + 8*half
    v8f Cp;
    const float* p2b = p2 + (size_t)b * N_;
#pragma unroll
    for (int r = 0; r < 8; ++r) Cp[r] = NEG_SCALE * p2b[i0 + r + 8 * half];

    float acc[8];
#pragma unroll
    for (int r = 0; r < 8; ++r) acc[r] = 0.f;

#pragma unroll 2
    for (int j0 = par * 16; j0 < N_; j0 += 32) {
        const int col = j0 + l16;
        float2 bb = qpan2[half][col];     // single ds_load_b64, branchless
        v2f Bm; Bm.x = bb.x; Bm.y = bb.y;
        v8f S = __builtin_amdgcn_wmma_f32_16x16x4_f32(
            false, A, false, Bm, (short)0, Cp, false, false);  // D = s directly
        const float wv = wls[col];
#pragma unroll
        for (int r = 0; r < 8; ++r) {
            float s = fminf(S[r], 0.0f);                 // clamp C >= 0
            acc[r] += __builtin_amdgcn_exp2f(s) * wv;
        }
    }

    // combine the wave pair through LDS
    if (par == 0) {
#pragma unroll
        for (int r = 0; r < 8; ++r) part[tile][r][lane] = acc[r];
    }
    __syncthreads();
    if (par == 1) {
#pragma unroll
        for (int r = 0; r < 8; ++r) {
            float s = acc[r] + part[tile][r][lane];
            s += __shfl_xor(s, 1, 32);
            s += __shfl_xor(s, 2, 32);
            s += __shfl_xor(s, 4, 32);
            s += __shfl_xor(s, 8, 32);
            if (l16 == 0) {
                int row = i0 + r + 8 * half;
                wout[(size_t)b * N_ + row] = INVN / (s + 1e-8f);
            }
        }
    }
}

// ---------------------------------------------------------------------------
// Final:  out += (1/B) * sum_{n,m} u[n] * K(n,m) * C(n,m) * v[m]
// Same WMMA folding with unscaled operands:
//   A = [-2x0, -2x1, -2x2, 1], B = [y0, y1, y2, y2norm], C = x2  =>  D = t.
// ---------------------------------------------------------------------------
__global__ __launch_bounds__(256) void emd_final(
    const float* __restrict__ x,  const float* __restrict__ y,
    const float* __restrict__ x2, const float* __restrict__ y2,
    const float* __restrict__ u,  const float* __restrict__ v,
    float* __restrict__ out)
{
    __shared__ float2 qpan2[2][N_];
    __shared__ float  wls[N_];   // v

    const int b   = blockIdx.x >> 5;
    const int grp = blockIdx.x & 31;
    const int tid = threadIdx.x;

    stage_panel(y + (size_t)b * D_ * N_, y2 + (size_t)b * N_, 1.0f,
                v + (size_t)b * N_, qpan2, wls, tid);
    __syncthreads();

    const int wave = tid >> 5;
    const int lane = tid & 31;
    const int half = lane >> 4;
    const int l16  = lane & 15;
    const int tile = wave >> 1;
    const int par  = wave & 1;
    const int i0   = (grp * 4 + tile) * 16;

    const float* Pb = x + (size_t)b * D_ * N_;
    const int arow = i0 + l16;
    v2f A;
    A.x = -2.0f * Pb[(2 * half) * N_ + arow];
    A.y = (half == 0) ? (-2.0f * Pb[1 * N_ + arow]) : 1.0f;

    v8f Cp;                       // x2 row norms
    float ur[8];
    const float* p2b = x2 + (size_t)b * N_;
    const float* ub  = u  + (size_t)b * N_;
#pragma unroll
    for (int r = 0; r < 8; ++r) {
        Cp[r] = p2b[i0 + r + 8 * half];
        ur[r] = ub [i0 + r + 8 * half];
    }

    float acc = 0.f;
#pragma unroll 2
    for (int j0 = par * 16; j0 < N_; j0 += 32) {
        const int col = j0 + l16;
        float2 bb = qpan2[half][col];
        v2f Bm; Bm.x = bb.x; Bm.y = bb.y;
        v8f T = __builtin_amdgcn_wmma_f32_16x16x4_f32(
            false, A, false, Bm, (short)0, Cp, false, false);  // D = cost t
        const float vv = wls[col];
#pragma unroll
        for (int r = 0; r < 8; ++r) {
            float t = fmaxf(T[r], 0.0f);
            float k = __builtin_amdgcn_exp2f(t * NEG_SCALE);
            acc += (ur[r] * vv) * (k * t);
        }
    }

    acc += __shfl_xor(acc, 1, 32);
    acc += __shfl_xor(acc, 2, 32);
    acc += __shfl_xor(acc, 4, 32);
    acc += __shfl_xor(acc, 8, 32);
    acc += __shfl_xor(acc, 16, 32);
    if (lane == 0) atomicAdd(out, acc * (1.0f / (float)B_));
}

// ---------------------------------------------------------------------------
extern "C" void kernel_launch(void* const* d_in, const int* in_sizes, int n_in,
                              void* d_out, int out_size, void* d_ws, size_t ws_size,
                              hipStream_t stream) {
    const float* x = (const float*)d_in[0];   // (16, 3, 2048)
    const float* y = (const float*)d_in[1];   // (16, 3, 2048)
    float* out = (float*)d_out;               // scalar f32
    float* ws  = (float*)d_ws;

    float* x2 = ws;                 // B*N
    float* y2 = ws + 1 * B_ * N_;   // B*N
    float* u  = ws + 2 * B_ * N_;   // B*N
    float* v  = ws + 3 * B_ * N_;   // B*N   (total 512 KB)

    emd_setup<<<(B_ * N_) / 256, 256, 0, stream>>>(x, y, x2, y2, u, v, out);

    dim3 grid(B_ * 32);   // 512 blocks, 4096 waves per pass
    for (int it = 0; it < ITERS; ++it) {
        // v = invN / (K^T u + 1e-8)   (dot product is symmetric -> swap roles)
        sinkhorn_matvec<<<grid, 256, 0, stream>>>(y, x, y2, x2, u, v);
        // u = invN / (K v + 1e-8)
        sinkhorn_matvec<<<grid, 256, 0, stream>>>(x, y, x2, y2, v, u);
    }

    emd_final<<<grid, 256, 0, stream>>>(x, y, x2, y2, u, v, out);
}